// GATEncoder_62508954026342
// MI455X (gfx1250) — compile-verified
//
#include <hip/hip_runtime.h>

// ---------------- problem constants (match reference) ----------------
#define NNODES  50000
#define NEDGES  800000
#define FDIM    256        // IN = HEADS*HID = OUT = 256
#define HEADS1  4
#define NEG_SLOPE 0.2f

typedef __attribute__((ext_vector_type(2))) float v2f;
typedef __attribute__((ext_vector_type(8))) float v8f;

// ---------------- ordered-uint encoding for float atomic max ----------------
__device__ __forceinline__ unsigned f2ord(float f) {
  unsigned u = __float_as_uint(f);
  return (u & 0x80000000u) ? ~u : (u | 0x80000000u);
}
__device__ __forceinline__ float ord2f(unsigned k) {
  unsigned u = (k & 0x80000000u) ? (k ^ 0x80000000u) : ~k;
  return __uint_as_float(u);
}
// f2ord(-inf) == 0x007FFFFF
#define ORD_NEG_INF 0x007FFFFFu

// ---------------- generic fill ----------------
__global__ void fill_u32(unsigned* __restrict__ p, long long count, unsigned val) {
  long long i = (long long)blockIdx.x * blockDim.x + threadIdx.x;
  if (i < count) p[i] = val;
}

// ---------------- fp32 GEMM via V_WMMA_F32_16X16X4_F32 ----------------
// C[M,256] = A[M,256] @ B[256,256].
// One wave32 computes a 16x64 output strip: 4 N-tiles sharing each A fragment
// (4 independent WMMA chains, A traffic /4, all offsets compile-time imms).
__global__ void gemm_wmma_f32(const float* __restrict__ A, const float* __restrict__ B,
                              float* __restrict__ C, int M) {
  constexpr int K = 256, Nc = 256;
  constexpr int NT = 4;                 // N-tiles per wave
  constexpr int nWavesN = (Nc / 16) / NT; // 4
  int wave = (int)(((long long)blockIdx.x * blockDim.x + threadIdx.x) >> 5);
  int lane = threadIdx.x & 31;
  int tilesM = M >> 4;
  if (wave >= tilesM * nWavesN) return;
  int tM  = wave / nWavesN;
  int tN4 = (wave - tM * nWavesN) * NT; // starting N-tile index
  int half = lane >> 4;                 // 0: K=k..k+1   1: K=k+2..k+3
  int l16  = lane & 15;                 // row (A) / col (B,C)

  const float* Abase = A + (size_t)(tM * 16 + l16) * K + 2 * half;
  const float* Bbase = B + (size_t)(2 * half) * Nc + tN4 * 16 + l16;

  v8f c[NT] = {};
#pragma unroll 8
  for (int k = 0; k < K; k += 4) {
    v2f a = *(const v2f*)(Abase + k);   // A[row][k+2h .. k+2h+1]
#pragma unroll
    for (int t = 0; t < NT; ++t) {
      v2f b;
      b.x = Bbase[k * Nc + t * 16];          // B[k+2h  ][col]
      b.y = Bbase[k * Nc + Nc + t * 16];     // B[k+2h+1][col]
      c[t] = __builtin_amdgcn_wmma_f32_16x16x4_f32(false, a, false, b,
                                                   (short)0, c[t], false, false);
    }
  }
  // C/D layout: VGPR i -> row tM*16 + i + 8*half, col = l16
  float* Cp = C + (size_t)(tM * 16 + 8 * half) * Nc + tN4 * 16 + l16;
#pragma unroll
  for (int t = 0; t < NT; ++t)
#pragma unroll
    for (int i = 0; i < 8; ++i) Cp[i * Nc + t * 16] = c[t][i];
}

// ---------------- per-node attention logits (fused src/dst dots) ----------------
// als[n,h] = <h[n,h,:], a_src[h,:]>, ald[n,h] = <h[n,h,:], a_dst[h,:]>
template <int H>
__global__ void att_logits(const float* __restrict__ h, const float* __restrict__ a_src,
                           const float* __restrict__ a_dst, float* __restrict__ als,
                           float* __restrict__ ald, int n) {
  constexpr int F = FDIM / H;
  int idx = blockIdx.x * blockDim.x + threadIdx.x;
  if (idx >= n * H) return;
  int node = idx / H, hd = idx - node * H;
  const float4* hp = (const float4*)(h + (size_t)node * FDIM + (size_t)hd * F);
  const float4* as = (const float4*)(a_src + hd * F);
  const float4* ad = (const float4*)(a_dst + hd * F);
  float s = 0.f, d = 0.f;
#pragma unroll 4
  for (int f = 0; f < F / 4; ++f) {
    float4 v = hp[f], x = as[f], y = ad[f];
    s += v.x * x.x + v.y * x.y + v.z * x.z + v.w * x.w;
    d += v.x * y.x + v.y * y.y + v.z * y.z + v.w * y.w;
  }
  als[idx] = s;
  ald[idx] = d;
}

template <int H>
__device__ __forceinline__ float edge_score(const float* als, const float* ald,
                                            int s, int d, int hd) {
  float sc = als[s * H + hd] + ald[d * H + hd];
  return sc > 0.f ? sc : NEG_SLOPE * sc;
}

// pass 1: segment max over dst (ordered-uint atomic max)
template <int H>
__global__ void edge_max(const int* __restrict__ src, const int* __restrict__ dst,
                         int E0, int n, const float* __restrict__ als,
                         const float* __restrict__ ald, unsigned* __restrict__ mx) {
  long long i = (long long)blockIdx.x * blockDim.x + threadIdx.x;
  long long ET = (long long)E0 + n;
  if (i >= ET * H) return;
  int hd = (int)(i % H);
  long long ed = i / H;
  int s = ed < E0 ? src[ed] : (int)(ed - E0);
  int d = ed < E0 ? dst[ed] : (int)(ed - E0);
  float sc = edge_score<H>(als, ald, s, d, hd);
  atomicMax(&mx[d * H + hd], f2ord(sc));
}

// pass 2: segment sum of exp(e - m[dst])
template <int H>
__global__ void edge_expsum(const int* __restrict__ src, const int* __restrict__ dst,
                            int E0, int n, const float* __restrict__ als,
                            const float* __restrict__ ald, const unsigned* __restrict__ mx,
                            float* __restrict__ dn) {
  long long i = (long long)blockIdx.x * blockDim.x + threadIdx.x;
  long long ET = (long long)E0 + n;
  if (i >= ET * H) return;
  int hd = (int)(i % H);
  long long ed = i / H;
  int s = ed < E0 ? src[ed] : (int)(ed - E0);
  int d = ed < E0 ? dst[ed] : (int)(ed - E0);
  float sc = edge_score<H>(als, ald, s, d, hd);
  float ex = __expf(sc - ord2f(mx[d * H + hd]));
  atomicAdd(&dn[d * H + hd], ex);
}

// pass 3: agg[dst] += alpha * h[src].
// 4 edges per 256-thread block, 64 threads per edge, float4 (b128) gathers.
template <int H>
__global__ void edge_agg(const int* __restrict__ src, const int* __restrict__ dst,
                         int E0, int n, const float* __restrict__ als,
                         const float* __restrict__ ald, const unsigned* __restrict__ mx,
                         const float* __restrict__ dn, const float* __restrict__ h,
                         float* __restrict__ agg) {
  long long ET = (long long)E0 + n;
  long long ed = (long long)blockIdx.x * 4 + (threadIdx.x >> 6);
  if (ed >= ET) return;
  int q = threadIdx.x & 63;            // features 4q..4q+3 (same head: blocks 64-aligned)
  int s = ed < E0 ? src[ed] : (int)(ed - E0);
  int d = ed < E0 ? dst[ed] : (int)(ed - E0);
  int hd = (q * H) >> 6;               // (4q) / (FDIM/H)
  float sc = edge_score<H>(als, ald, s, d, hd);
  float ex = __expf(sc - ord2f(mx[d * H + hd]));
  float alpha = ex / (dn[d * H + hd] + 1e-16f);
  float4 v = ((const float4*)(h + (size_t)s * FDIM))[q];
  float* ap = agg + (size_t)d * FDIM + 4 * q;
  atomicAdd(ap + 0, v.x * alpha);
  atomicAdd(ap + 1, v.y * alpha);
  atomicAdd(ap + 2, v.z * alpha);
  atomicAdd(ap + 3, v.w * alpha);
}

// layer-1 finalize: x = elu(agg + b)
__global__ void bias_elu(float* __restrict__ agg, const float* __restrict__ b,
                         long long total) {
  long long i = (long long)blockIdx.x * blockDim.x + threadIdx.x;
  if (i >= total) return;
  float v = agg[i] + b[i & (FDIM - 1)];
  agg[i] = v > 0.f ? v : expm1f(v);
}

// layer-2 finalize: out += b  (heads==1 -> mean over 1 head is identity)
__global__ void bias_add(float* __restrict__ out, const float* __restrict__ b,
                         long long total) {
  long long i = (long long)blockIdx.x * blockDim.x + threadIdx.x;
  if (i >= total) return;
  out[i] += b[i & (FDIM - 1)];
}

static inline int blks(long long n, int bs) { return (int)((n + bs - 1) / bs); }

extern "C" void kernel_launch(void* const* d_in, const int* in_sizes, int n_in,
                              void* d_out, int out_size, void* d_ws, size_t ws_size,
                              hipStream_t stream) {
  const float* x        = (const float*)d_in[0];
  const int*   ei       = (const int*)d_in[1];
  const float* W1       = (const float*)d_in[2];
  const float* att_src1 = (const float*)d_in[3];
  const float* att_dst1 = (const float*)d_in[4];
  const float* b1       = (const float*)d_in[5];
  const float* W2       = (const float*)d_in[6];
  const float* att_src2 = (const float*)d_in[7];
  const float* att_dst2 = (const float*)d_in[8];
  const float* b2       = (const float*)d_in[9];
  float* out = (float*)d_out;

  const int* src = ei;               // edge_index[0,:]
  const int* dst = ei + NEDGES;      // edge_index[1,:]

  const long long NF = (long long)NNODES * FDIM;   // 12.8M floats
  const long long NH = (long long)NNODES * HEADS1; // 200K

  float* ws = (float*)d_ws;
  float*    h    = ws;                    // [N,256] dense GEMM output
  float*    agg  = ws + NF;               // [N,256] aggregation / layer-2 input
  float*    als  = ws + 2 * NF;           // [N,H]
  float*    ald  = als + NH;              // [N,H]
  unsigned* mx   = (unsigned*)(ald + NH); // [N,H] ordered-uint max
  float*    dn   = (float*)(mx + NH);     // [N,H] softmax denom
  (void)ws_size; (void)in_sizes; (void)n_in; (void)out_size;

  const long long ET = (long long)NEDGES + NNODES; // with self loops
  const long long gemmThreads = (long long)(NNODES / 16) * 4 /*nWavesN*/ * 32;
  const int aggBlks = blks(ET, 4);

  // ---------------- layer 1 (H = 4) ----------------
  gemm_wmma_f32<<<blks(gemmThreads, 256), 256, 0, stream>>>(x, W1, h, NNODES);
  att_logits<HEADS1><<<blks(NH, 256), 256, 0, stream>>>(h, att_src1, att_dst1,
                                                        als, ald, NNODES);
  fill_u32<<<blks(NF, 256), 256, 0, stream>>>((unsigned*)agg, NF, 0u);
  fill_u32<<<blks(NH, 256), 256, 0, stream>>>(mx, NH, ORD_NEG_INF);
  fill_u32<<<blks(NH, 256), 256, 0, stream>>>((unsigned*)dn, NH, 0u);
  edge_max<HEADS1><<<blks(ET * HEADS1, 256), 256, 0, stream>>>(src, dst, NEDGES, NNODES,
                                                               als, ald, mx);
  edge_expsum<HEADS1><<<blks(ET * HEADS1, 256), 256, 0, stream>>>(src, dst, NEDGES, NNODES,
                                                                  als, ald, mx, dn);
  edge_agg<HEADS1><<<aggBlks, 256, 0, stream>>>(src, dst, NEDGES, NNODES,
                                                als, ald, mx, dn, h, agg);
  bias_elu<<<blks(NF, 256), 256, 0, stream>>>(agg, b1, NF);

  // ---------------- layer 2 (H = 1) ----------------
  gemm_wmma_f32<<<blks(gemmThreads, 256), 256, 0, stream>>>(agg, W2, h, NNODES);
  att_logits<1><<<blks(NNODES, 256), 256, 0, stream>>>(h, att_src2, att_dst2,
                                                       als, ald, NNODES);
  fill_u32<<<blks(NF, 256), 256, 0, stream>>>((unsigned*)out, NF, 0u);
  fill_u32<<<blks(NNODES, 256), 256, 0, stream>>>(mx, NNODES, ORD_NEG_INF);
  fill_u32<<<blks(NNODES, 256), 256, 0, stream>>>((unsigned*)dn, NNODES, 0u);
  edge_max<1><<<blks(ET, 256), 256, 0, stream>>>(src, dst, NEDGES, NNODES, als, ald, mx);
  edge_expsum<1><<<blks(ET, 256), 256, 0, stream>>>(src, dst, NEDGES, NNODES,
                                                    als, ald, mx, dn);
  edge_agg<1><<<aggBlks, 256, 0, stream>>>(src, dst, NEDGES, NNODES,
                                           als, ald, mx, dn, h, out);
  bias_add<<<blks(NF, 256), 256, 0, stream>>>(out, b2, NF);
}